// BeamSearchAttentionDecoder_55181739819093
// MI455X (gfx1250) — compile-verified
//
#include <hip/hip_runtime.h>
#include <math.h>
#include <stdint.h>

#define HDIM   1024
#define SDIM   512
#define TOPKV  5
#define BATCHV 160
#define NSTEPS 4
#define NBASE  (BATCHV / TOPKV)   // 32
#define NEGVAL -10000000000.0f
#define RSQRT_H 0.03125f          // 1/sqrt(1024)

typedef __attribute__((ext_vector_type(2))) float v2f;
typedef __attribute__((ext_vector_type(8))) float v8f;

// Async global -> LDS copy (16B per lane), tracked by ASYNCcnt (CDNA5 TDM-adjacent
// path; inline asm is portable across ROCm 7.2 / amdgpu-toolchain builtins).
#define ASYNC_CP_B128(ldsoff, gptr)                                     \
  asm volatile("global_load_async_to_lds_b128 %0, %1, off"              \
               :: "v"(ldsoff), "v"(gptr) : "memory")
#define WAIT_ASYNC0() asm volatile("s_wait_asynccnt 0x0" ::: "memory")

// ---------------------------------------------------------------------------
// WMMA f32 GEMM with double-buffered async-LDS staging.
//   C[M,N] = A[M,K] @ op(B) + bias
//   TRANSB=true :  op(B)=B^T, B stored (N,K)  ("x @ W.T")
//   TRANSB=false:  op(B)=B,   B stored (K,N)
// Block = 256 threads = 8 waves -> 32x64 C tile (2x4 waves of 16x16).
// K-tile = 64. LDS: A 2x32x68, B 2x64x68 (pad 68 -> bank-conflict-free frags).
// All call-site M,N,K are exact multiples -> no predication, EXEC all-ones at
// every v_wmma (ISA requirement).
// ---------------------------------------------------------------------------
template <bool TRANSB>
__global__ __launch_bounds__(256) void gemm_wmma_lds(
    const float* __restrict__ A, const float* __restrict__ B,
    const float* __restrict__ bias, float* __restrict__ C,
    int M, int N, int K) {
  constexpr int KT  = 64;
  constexpr int LDA = 68;
  __shared__ float Ash[2][32 * LDA];
  __shared__ float Bsh[2][64 * LDA];

  const int tid  = threadIdx.x;
  const int lane = tid & 31;
  const int wave = tid >> 5;
  const int wm = wave >> 2, wn = wave & 3;
  const int m0 = blockIdx.y * 32;
  const int n0 = blockIdx.x * 64;
  const int wave_m0 = m0 + wm * 16;
  const int wave_n0 = n0 + wn * 16;

  // ---- staging coordinates: thread t fills row sr (and sr+16[,+32,+48]) at
  //      column group sc (4 floats = one b128) of the K-tile panel.
  const int sr = tid >> 4;         // 0..15
  const int sc = (tid & 15) * 4;   // 0..60

  const float* gA0 = A + (size_t)(m0 + sr) * K + sc;
  const float* gA1 = A + (size_t)(m0 + sr + 16) * K + sc;
  const float *gB0, *gB1, *gB2, *gB3;
  if (TRANSB) {   // panel rows are N
    gB0 = B + (size_t)(n0 + sr) * K + sc;
    gB1 = B + (size_t)(n0 + sr + 16) * K + sc;
    gB2 = B + (size_t)(n0 + sr + 32) * K + sc;
    gB3 = B + (size_t)(n0 + sr + 48) * K + sc;
  } else {        // panel rows are K
    gB0 = B + (size_t)(sr) * N + n0 + sc;
    gB1 = B + (size_t)(sr + 16) * N + n0 + sc;
    gB2 = B + (size_t)(sr + 32) * N + n0 + sc;
    gB3 = B + (size_t)(sr + 48) * N + n0 + sc;
  }

  // LDS byte offsets (low 32 bits of a generic shared pointer == LDS offset)
  const unsigned ldsA = (unsigned)(uintptr_t)&Ash[0][0];
  const unsigned ldsB = (unsigned)(uintptr_t)&Bsh[0][0];
  const unsigned aoff = ldsA + (unsigned)((sr * LDA + sc) * 4);
  const unsigned boff = ldsB + (unsigned)((sr * LDA + sc) * 4);
  constexpr unsigned ROW16 = 16u * LDA * 4u;
  constexpr unsigned ABUF  = 32u * LDA * 4u;
  constexpr unsigned BBUF  = 64u * LDA * 4u;

  const int ntiles = K / KT;

  auto stage = [&](int buf, int t) {
    const size_t ka = (size_t)t * KT;
    const size_t kb = TRANSB ? ka : (size_t)t * KT * N;
    const unsigned ab = buf ? ABUF : 0u;
    const unsigned bb = buf ? BBUF : 0u;
    ASYNC_CP_B128(aoff + ab,             gA0 + ka);
    ASYNC_CP_B128(aoff + ROW16 + ab,     gA1 + ka);
    ASYNC_CP_B128(boff + bb,             gB0 + kb);
    ASYNC_CP_B128(boff + ROW16 + bb,     gB1 + kb);
    ASYNC_CP_B128(boff + 2 * ROW16 + bb, gB2 + kb);
    ASYNC_CP_B128(boff + 3 * ROW16 + bb, gB3 + kb);
  };

  stage(0, 0);
  WAIT_ASYNC0();
  __syncthreads();

  // ---- WMMA fragment coordinates (ISA 16x4 f32 A / 4x16 B layouts)
  const int fr = lane & 15;          // M (A) / N (B) within tile
  const int kh = (lane >> 4) << 1;   // K sub-offset {0,1} vs {2,3}

  v8f acc = {};
  int buf = 0;
  for (int t = 0; t < ntiles; ++t) {
    if (t + 1 < ntiles) stage(buf ^ 1, t + 1);

    const float* Asp = &Ash[buf][(wm * 16 + fr) * LDA + kh];
    const float* Bsp = TRANSB ? &Bsh[buf][(wn * 16 + fr) * LDA + kh] : &Bsh[buf][0];
#pragma unroll
    for (int k = 0; k < KT; k += 4) {
      v2f a = *(const v2f*)(Asp + k);
      v2f b;
      if (TRANSB) {
        b = *(const v2f*)(Bsp + k);
      } else {
        b.x = Bsh[buf][(k + kh) * LDA + wn * 16 + fr];
        b.y = Bsh[buf][(k + kh + 1) * LDA + wn * 16 + fr];
      }
      acc = __builtin_amdgcn_wmma_f32_16x16x4_f32(false, a, false, b,
                                                  (short)0, acc, false, false);
    }
    WAIT_ASYNC0();
    __syncthreads();
    buf ^= 1;
  }

  const int col = wave_n0 + fr;
  const int rb  = (lane >> 4) * 8;   // C layout: vgpr i -> M=i / M=i+8
  const float bv = bias ? bias[col] : 0.0f;
#pragma unroll
  for (int i = 0; i < 8; i++) {
    C[(size_t)(wave_m0 + rb + i) * N + col] = acc[i] + bv;
  }
}

// ---------------------------------------------------------------------------
// GRU pointwise gates: h' = (1-z)*n + z*h
// ---------------------------------------------------------------------------
__global__ void gru_pointwise_k(const float* __restrict__ gi,
                                const float* __restrict__ gh,
                                const float* __restrict__ h_prev,
                                float* __restrict__ h_new) {
  int idx = blockIdx.x * blockDim.x + threadIdx.x;  // BATCH*H
  int b = idx >> 10, j = idx & (HDIM - 1);
  const float* gib = gi + (size_t)b * 3 * HDIM;
  const float* ghb = gh + (size_t)b * 3 * HDIM;
  float r = 1.0f / (1.0f + __expf(-(gib[j] + ghb[j])));
  float z = 1.0f / (1.0f + __expf(-(gib[HDIM + j] + ghb[HDIM + j])));
  float n = tanhf(gib[2 * HDIM + j] + r * ghb[2 * HDIM + j]);
  h_new[idx] = (1.0f - z) * n + z * h_prev[idx];
}

// hb1[b] = dot(h_new[b,:], b1)  (one wave per row)
__global__ void dot_b1_k(const float* __restrict__ h_new,
                         const float* __restrict__ b1,
                         float* __restrict__ hb1) {
  int wv = (blockIdx.x * blockDim.x + threadIdx.x) >> 5;
  int lane = threadIdx.x & 31;
  if (wv >= BATCHV) return;
  float s = 0.0f;
#pragma unroll 8
  for (int j = lane; j < HDIM; j += 32) s += h_new[(size_t)wv * HDIM + j] * b1[j];
  for (int o = 16; o > 0; o >>= 1) s += __shfl_down(s, o, 32);
  if (lane == 0) hb1[wv] = s;
}

// ---------------------------------------------------------------------------
// attn_out[b,s] = (q[b]·enc[b,s] + h·b1)/sqrt(H) + mask[b,s]
// Stream 1 over enc (335 MB) with b128 loads; one wave per (b,s).
// ---------------------------------------------------------------------------
__global__ __launch_bounds__(256) void attn_scores_k(
    const float* __restrict__ q, const float* __restrict__ enc,
    const float* __restrict__ hb1, const float* __restrict__ mask,
    float* __restrict__ attn_out) {
  int b = blockIdx.y;
  int s = blockIdx.x * 8 + (threadIdx.x >> 5);
  int lane = threadIdx.x & 31;
  const float* qb = q + (size_t)b * HDIM;
  const float* eb = enc + ((size_t)b * SDIM + s) * HDIM;
  if (s + 8 < SDIM) __builtin_prefetch(eb + 8 * HDIM + lane * 4, 0, 1);
  float sum = 0.0f;
#pragma unroll
  for (int j = lane * 4; j < HDIM; j += 128) {
    float4 qv = *(const float4*)(qb + j);
    float4 ev = *(const float4*)(eb + j);
    sum += qv.x * ev.x + qv.y * ev.y + qv.z * ev.z + qv.w * ev.w;
  }
  for (int o = 16; o > 0; o >>= 1) sum += __shfl_down(sum, o, 32);
  if (lane == 0)
    attn_out[b * SDIM + s] = (sum + hb1[b]) * RSQRT_H + mask[b * SDIM + s];
}

// ---------------------------------------------------------------------------
// Softmax over S plus top-5 (values desc, stable smallest-index ties).
// ---------------------------------------------------------------------------
__global__ __launch_bounds__(512) void softmax_topk_k(
    const float* __restrict__ attn_out, float* __restrict__ alpha,
    float* __restrict__ scores, int* __restrict__ sent) {
  int b = blockIdx.x, t = threadIdx.x;
  __shared__ float rv[SDIM];
  __shared__ int   ri[SDIM];
  float x = attn_out[b * SDIM + t];
  rv[t] = x; __syncthreads();
  for (int o = 256; o > 0; o >>= 1) { if (t < o) rv[t] = fmaxf(rv[t], rv[t + o]); __syncthreads(); }
  float m = rv[0]; __syncthreads();
  float e = __expf(x - m);
  rv[t] = e; __syncthreads();
  for (int o = 256; o > 0; o >>= 1) { if (t < o) rv[t] += rv[t + o]; __syncthreads(); }
  float denom = rv[0]; __syncthreads();
  float a = e / denom;
  alpha[b * SDIM + t] = a;

  float mya = a;
  for (int k = 0; k < TOPKV; k++) {
    rv[t] = mya; ri[t] = t; __syncthreads();
    for (int o = 256; o > 0; o >>= 1) {
      if (t < o) {
        if (rv[t + o] > rv[t] || (rv[t + o] == rv[t] && ri[t + o] < ri[t])) {
          rv[t] = rv[t + o]; ri[t] = ri[t + o];
        }
      }
      __syncthreads();
    }
    if (t == 0) { scores[b * TOPKV + k] = rv[0]; sent[b * TOPKV + k] = ri[0]; }
    __syncthreads();
    if (t == ri[0]) mya = -3.4e38f;
    __syncthreads();
  }
}

// ---------------------------------------------------------------------------
// Stream 2 over enc: partial ctx over 4 S-chunks (640 blocks for concurrency),
// each thread owns 4 H-columns (b128 loads), then a tiny reduce.
// ---------------------------------------------------------------------------
__global__ __launch_bounds__(256) void ctx_part_k(const float* __restrict__ alpha,
                                                  const float* __restrict__ enc,
                                                  float* __restrict__ part) {
  int b = blockIdx.y, p = blockIdx.x;            // p: chunk of 128 s-values
  int h4 = threadIdx.x * 4;
  const float* ab = alpha + b * SDIM + p * 128;
  const float* eb = enc + ((size_t)b * SDIM + p * 128) * HDIM + h4;
  float4 acc = {0.f, 0.f, 0.f, 0.f};
#pragma unroll 4
  for (int s = 0; s < 128; ++s) {
    float w = ab[s];
    float4 ev = *(const float4*)(eb + (size_t)s * HDIM);
    acc.x += w * ev.x; acc.y += w * ev.y; acc.z += w * ev.z; acc.w += w * ev.w;
  }
  *(float4*)(part + ((size_t)(b * 4 + p)) * HDIM + h4) = acc;
}

__global__ void ctx_reduce_k(const float* __restrict__ part,
                             float* __restrict__ ctx_raw) {
  int idx = blockIdx.x * blockDim.x + threadIdx.x;  // BATCH*H
  int b = idx >> 10, h = idx & (HDIM - 1);
  const float* pb = part + (size_t)b * 4 * HDIM + h;
  ctx_raw[idx] = pb[0] + pb[HDIM] + pb[2 * HDIM] + pb[3 * HDIM];
}

// cat = [context | h_new]  (160 x 2048)
__global__ void concat_k(const float* __restrict__ ctx,
                         const float* __restrict__ h_new,
                         float* __restrict__ cat) {
  int idx = blockIdx.x * blockDim.x + threadIdx.x;  // BATCH*2H
  int b = idx >> 11, j = idx & (2 * HDIM - 1);
  cat[idx] = (j < HDIM) ? ctx[b * HDIM + j] : h_new[b * HDIM + (j - HDIM)];
}

// ---------------------------------------------------------------------------
// Beam reorder: per base (32), 5 smallest of 25 (-log(score)+evidence),
// stable ascending like jnp.argsort.
// ---------------------------------------------------------------------------
__global__ void beam_k(const float* __restrict__ scores,
                       const int* __restrict__ sent,
                       const float* __restrict__ evidence,
                       int* __restrict__ s_flat, float* __restrict__ newev,
                       int* __restrict__ newsent) {
  int base = threadIdx.x;
  if (base >= NBASE) return;
  float ls[TOPKV * TOPKV];
  bool used[TOPKV * TOPKV];
#pragma unroll
  for (int i = 0; i < TOPKV * TOPKV; i++) {
    int row = base * TOPKV + i / TOPKV;
    int rr = i % TOPKV;
    ls[i] = -__logf(scores[row * TOPKV + rr]) + evidence[row];
    used[i] = false;
  }
  for (int k = 0; k < TOPKV; k++) {
    int best = -1; float bv = 3.4e38f;
#pragma unroll
    for (int i = 0; i < TOPKV * TOPKV; i++)
      if (!used[i] && ls[i] < bv) { bv = ls[i]; best = i; }
    used[best] = true;
    int row = base * TOPKV + best / TOPKV;
    int rr = best % TOPKV;
    int out = base * TOPKV + k;
    s_flat[out] = row;
    newev[out] = bv;
    newsent[out] = sent[row * TOPKV + rr];
  }
}

// ---------------------------------------------------------------------------
// Assemble outputs (concatenated flat, reference return order).
// ---------------------------------------------------------------------------
#define O1 0
#define O2 (BATCHV * HDIM)                       // hidden_new
#define O3 (2 * BATCHV * HDIM)                   // ev_idx_new
#define O4 (O3 + BATCHV * TOPKV)                 // attn_scores_new
#define O5 (O4 + (NSTEPS + 1) * BATCHV * SDIM)   // mask_new
#define O6 (O5 + BATCHV * SDIM)                  // new_evidence_scores

__global__ __launch_bounds__(256) void assemble_k(
    const float* __restrict__ result, const float* __restrict__ h_new,
    const int* __restrict__ ev_idx_in, const float* __restrict__ attn_scores_in,
    const float* __restrict__ mask_in, const float* __restrict__ attn_out,
    const int* __restrict__ s_flat, const int* __restrict__ newsent,
    const float* __restrict__ newev, float* __restrict__ out) {
  int i = blockIdx.x, t = threadIdx.x;
  int src = s_flat[i];
  int ns = newsent[i];
  for (int j = t; j < HDIM; j += 256) {
    out[O1 + i * HDIM + j] = result[(size_t)src * HDIM + j];
    out[O2 + i * HDIM + j] = h_new[(size_t)src * HDIM + j];
  }
  if (t < NSTEPS) out[O3 + i * TOPKV + t] = (float)ev_idx_in[src * NSTEPS + t];
  if (t == NSTEPS) out[O3 + i * TOPKV + NSTEPS] = (float)ns;
  for (int j = t; j < SDIM; j += 256) {
#pragma unroll
    for (int st = 0; st < NSTEPS; st++)
      out[O4 + (st * BATCHV + i) * SDIM + j] =
          attn_scores_in[(st * BATCHV + src) * SDIM + j];
    out[O4 + (NSTEPS * BATCHV + i) * SDIM + j] = attn_out[src * SDIM + j];
    float mv = mask_in[src * SDIM + j];
    out[O5 + i * SDIM + j] = (j == ns) ? NEGVAL : mv;
  }
  if (t == 0) out[O6 + i] = newev[i];
}

// ---------------------------------------------------------------------------
extern "C" void kernel_launch(void* const* d_in, const int* in_sizes, int n_in,
                              void* d_out, int out_size, void* d_ws, size_t ws_size,
                              hipStream_t stream) {
  const float* last_hidden    = (const float*)d_in[0];   // (1,160,1024)
  const float* dec_in         = (const float*)d_in[1];   // (160,1,1024)
  const float* enc            = (const float*)d_in[2];   // (160,512,1024)
  const float* attn_scores_in = (const float*)d_in[3];   // (4,160,1,512)
  const float* mask           = (const float*)d_in[4];   // (160,1,512)
  const float* evidence       = (const float*)d_in[5];   // (160,)
  const int*   ev_idx_in      = (const int*)d_in[6];     // (160,4)
  const float* W1   = (const float*)d_in[7];
  const float* b1   = (const float*)d_in[8];
  const float* W2   = (const float*)d_in[9];
  const float* b2   = (const float*)d_in[10];
  const float* W3   = (const float*)d_in[11];
  const float* b3   = (const float*)d_in[12];
  const float* W_ih = (const float*)d_in[13];
  const float* W_hh = (const float*)d_in[14];
  const float* b_ih = (const float*)d_in[15];
  const float* b_hh = (const float*)d_in[16];

  float* ws       = (float*)d_ws;
  float* gi       = ws;                            // 160*3072
  float* gh       = gi + BATCHV * 3 * HDIM;        // 160*3072
  float* h_new    = gh + BATCHV * 3 * HDIM;        // 160*1024
  float* q        = h_new + BATCHV * HDIM;         // 160*1024
  float* hb1      = q + BATCHV * HDIM;             // 160
  float* attn_out = hb1 + BATCHV;                  // 160*512
  float* alpha    = attn_out + BATCHV * SDIM;      // 160*512
  float* ctx_part = alpha + BATCHV * SDIM;         // 160*4*1024
  float* ctx_raw  = ctx_part + BATCHV * 4 * HDIM;  // 160*1024
  float* ctx      = ctx_raw + BATCHV * HDIM;       // 160*1024
  float* cat      = ctx + BATCHV * HDIM;           // 160*2048
  float* result   = cat + BATCHV * 2 * HDIM;       // 160*1024
  float* scores   = result + BATCHV * HDIM;        // 160*5
  float* newev    = scores + BATCHV * TOPKV;       // 160
  int*   sent     = (int*)(newev + BATCHV);        // 160*5
  int*   s_flat   = sent + BATCHV * TOPKV;         // 160
  int*   newsent  = s_flat + BATCHV;               // 160

  dim3 blk(256);

  // GRU gates via WMMA + async-LDS double buffering
  gemm_wmma_lds<true><<<dim3(3 * HDIM / 64, BATCHV / 32), blk, 0, stream>>>(
      dec_in, W_ih, b_ih, gi, BATCHV, 3 * HDIM, HDIM);
  gemm_wmma_lds<true><<<dim3(3 * HDIM / 64, BATCHV / 32), blk, 0, stream>>>(
      last_hidden, W_hh, b_hh, gh, BATCHV, 3 * HDIM, HDIM);
  gru_pointwise_k<<<dim3(BATCHV * HDIM / 256), blk, 0, stream>>>(gi, gh, last_hidden, h_new);

  // q = h_new @ W1 (linearity refactor kills the 172-GF key_enc GEMM)
  gemm_wmma_lds<false><<<dim3(HDIM / 64, BATCHV / 32), blk, 0, stream>>>(
      h_new, W1, nullptr, q, BATCHV, HDIM, HDIM);
  dot_b1_k<<<dim3(BATCHV * 32 / 256), blk, 0, stream>>>(h_new, b1, hb1);

  // Stream 1 over enc: attention logits (+mask, pre-softmax; reused as output)
  attn_scores_k<<<dim3(SDIM / 8, BATCHV), blk, 0, stream>>>(q, enc, hb1, mask, attn_out);
  softmax_topk_k<<<dim3(BATCHV), dim3(SDIM), 0, stream>>>(attn_out, alpha, scores, sent);

  // Stream 2 over enc: ctx_raw = alpha @ enc ; context = ctx_raw @ W2.T + b2
  ctx_part_k<<<dim3(4, BATCHV), blk, 0, stream>>>(alpha, enc, ctx_part);
  ctx_reduce_k<<<dim3(BATCHV * HDIM / 256), blk, 0, stream>>>(ctx_part, ctx_raw);
  gemm_wmma_lds<true><<<dim3(HDIM / 64, BATCHV / 32), blk, 0, stream>>>(
      ctx_raw, W2, b2, ctx, BATCHV, HDIM, HDIM);

  // result = [context | h_new] @ W3.T + b3
  concat_k<<<dim3(BATCHV * 2 * HDIM / 256), blk, 0, stream>>>(ctx, h_new, cat);
  gemm_wmma_lds<true><<<dim3(HDIM / 64, BATCHV / 32), blk, 0, stream>>>(
      cat, W3, b3, result, BATCHV, HDIM, 2 * HDIM);

  // Beam reorder + output assembly
  beam_k<<<dim3(1), dim3(32), 0, stream>>>(scores, sent, evidence, s_flat, newev, newsent);
  assemble_k<<<dim3(BATCHV), blk, 0, stream>>>(result, h_new, ev_idx_in, attn_scores_in,
                                               mask, attn_out, s_flat, newsent, newev,
                                               (float*)d_out);
}